// GATLayer_34823594836461
// MI455X (gfx1250) — compile-verified
//
#include <hip/hip_runtime.h>
#include <math.h>

typedef float v2f __attribute__((ext_vector_type(2)));
typedef float v8f __attribute__((ext_vector_type(8)));

#define DFEAT 64
#define KNBR  30

__device__ __forceinline__ float softplusf(float x) {
    // jax.nn.softplus(x) = logaddexp(x, 0) = max(x,0) + log1p(exp(-|x|))
    return fmaxf(x, 0.0f) + log1pf(expf(-fabsf(x)));
}

// ---------------------------------------------------------------------------
// Kernel A: per-edge attention scores via V_WMMA_F32_16X16X4_F32 (exact f32).
// One wave32 computes 16 edge scores. A = 16 edges x 4 features (f32 A tile:
// lanes 0-15 hold K=0,1 in v0,v1; lanes 16-31 hold K=2,3). B = the matching
// 4 elements of W_attn replicated across all 16 columns, so every column of
// the 16x16 D accumulator equals the edge score. 32 chained WMMAs contract
// the full 128-feature dot product.
// ---------------------------------------------------------------------------
__global__ __launch_bounds__(256) void gat_edge_scores(
    const float* __restrict__ zf, const float* __restrict__ zo,
    const float* __restrict__ W, float* __restrict__ e, int E)
{
    const int lane = threadIdx.x & 31;
    const int wave = (int)((blockIdx.x * blockDim.x + threadIdx.x) >> 5);
    const long base = (long)wave * 16;
    if (base >= E) return;              // uniform per wave (E % 16 == 0)

    const int m  = lane & 15;           // edge row within 16x4 A tile
    const int kb = (lane >> 4) * 2;     // K pair base: 0 (lanes 0-15) or 2 (16-31)

    const float* zfr = zf + (base + m) * DFEAT;
    const float* zor = zo + (base + m) * DFEAT;

    v8f c = {0.f, 0.f, 0.f, 0.f, 0.f, 0.f, 0.f, 0.f};
    #pragma unroll
    for (int ch = 0; ch < 16; ++ch) {
        v2f a = *(const v2f*)(zfr + 4 * ch + kb);
        v2f b = *(const v2f*)(W   + 4 * ch + kb);   // replicate W chunk into all 16 cols
        c = __builtin_amdgcn_wmma_f32_16x16x4_f32(
                /*neg_a=*/false, a, /*neg_b=*/false, b,
                /*c_mod=*/(short)0, c, /*reuse_a=*/false, /*reuse_b=*/false);
    }
    #pragma unroll
    for (int ch = 0; ch < 16; ++ch) {
        v2f a = *(const v2f*)(zor + 4 * ch + kb);
        v2f b = *(const v2f*)(W + DFEAT + 4 * ch + kb);
        c = __builtin_amdgcn_wmma_f32_16x16x4_f32(
                false, a, false, b, (short)0, c, false, false);
    }

    // All 16 columns of D are identical. C/D layout: lane 0 holds rows M=0..7
    // (edges base+0..7) in c[0..7]; lane 16 holds rows M=8..15.
    if (m == 0) {
        float* dst = e + base + (lane >> 4) * 8;
        #pragma unroll
        for (int i = 0; i < 8; ++i) dst[i] = softplusf(c[i]);
    }
}

// ---------------------------------------------------------------------------
// Kernel B: per-node masked softmax over gathered scores + weighted feature
// aggregation. One wave32 per node; lane k owns neighbor slot k (K=30<=32).
// Each gathered z_others row is two fully-coalesced 128B wave reads.
// ---------------------------------------------------------------------------
__global__ __launch_bounds__(256) void gat_aggregate(
    const float* __restrict__ e, const float* __restrict__ zo,
    const int* __restrict__ scope, float* __restrict__ out, int N)
{
    const int lane = threadIdx.x & 31;
    const int node = (int)((blockIdx.x * blockDim.x + threadIdx.x) >> 5);
    if (node >= N) return;              // uniform per wave

    // Gather score for neighbor slot `lane`; scope is 1-based, 0 = padding.
    int   idx = (lane < KNBR) ? scope[(long)node * KNBR + lane] : 0;
    float ev  = (idx > 0) ? e[idx - 1] : 0.0f;
    bool  valid = (ev != 0.0f);         // reference: mask = (gathered e != 0)

    // Masked max over the wave
    float mx = valid ? ev : -INFINITY;
    #pragma unroll
    for (int off = 16; off > 0; off >>= 1)
        mx = fmaxf(mx, __shfl_xor(mx, off, 32));

    float p = valid ? expf(ev - mx) : 0.0f;
    float s = p;
    #pragma unroll
    for (int off = 16; off > 0; off >>= 1)
        s += __shfl_xor(s, off, 32);

    float alpha = (p > 0.0f) ? (p / s) : 0.0f;

    // out[node] = sum_k alpha_k * z_others[idx_k - 1]; lane owns features
    // d = lane and d = lane + 32.
    float acc0 = 0.0f, acc1 = 0.0f;
    for (int k = 0; k < KNBR; ++k) {
        float a  = __shfl(alpha, k, 32);
        int   id = __shfl(idx,   k, 32);
        if (a != 0.0f) {                // uniform branch (broadcast values)
            const float* row = zo + (long)(id - 1) * DFEAT;
            acc0 = fmaf(a, row[lane],      acc0);
            acc1 = fmaf(a, row[lane + 32], acc1);
        }
    }

    float* orow = out + (long)(node + 1) * DFEAT;   // row 0 is the zero row
    orow[lane]      = acc0;
    orow[lane + 32] = acc1;
}

__global__ void gat_zero_row0(float* __restrict__ out) {
    if (threadIdx.x < DFEAT) out[threadIdx.x] = 0.0f;
}

// ---------------------------------------------------------------------------
extern "C" void kernel_launch(void* const* d_in, const int* in_sizes, int n_in,
                              void* d_out, int out_size, void* d_ws, size_t ws_size,
                              hipStream_t stream) {
    const float* zf    = (const float*)d_in[0];   // [E, 64]
    const float* zo    = (const float*)d_in[1];   // [E, 64]
    const float* W     = (const float*)d_in[2];   // [128, 1]
    const int*   scope = (const int*)  d_in[3];   // [N, 30]

    const int E = in_sizes[0] / DFEAT;            // 900000
    const int N = in_sizes[3] / KNBR;             // 30000

    float* e   = (float*)d_ws;                    // E floats of scratch
    float* out = (float*)d_out;                   // [(N+1), 64]

    // Kernel A: one wave per 16 edges, 8 waves (256 threads) per block.
    const int tiles    = (E + 15) / 16;
    const int blocksA  = (tiles + 7) / 8;
    gat_edge_scores<<<blocksA, 256, 0, stream>>>(zf, zo, W, e, E);

    // Kernel B: one wave per node, 8 waves per block.
    const int blocksB = (N + 7) / 8;
    gat_aggregate<<<blocksB, 256, 0, stream>>>(e, zo, scope, out, N);

    gat_zero_row0<<<1, 64, 0, stream>>>(out);
}